// LIF_ODE_49185965474310
// MI455X (gfx1250) — compile-verified
//
#include <hip/hip_runtime.h>
#include <stdint.h>

// LIF network scan: T=35000 steps, B=256 batch, 8 pops, 12 connections.
// One lane per batch element; all state in VGPRs; 8 single-wave workgroups.
// Inputs streamed via the gfx1250 Tensor Data Mover: one tensor_load_to_lds
// per input per chunk (2D tile, LDS padding interleaves on/off planes),
// double-buffered and synchronized with s_wait_tensorcnt.
// Per-step math packed into V_PK_FMA_F32 pairs (pops & connections paired).

#define T_STEPS 35000
#define BATCH   256
#define CHUNK   100                     // timesteps per LDS chunk
#define NCHUNK  (T_STEPS / CHUNK)       // 350
#define BUF_BYTES (CHUNK * 2 * 32 * 4)  // 25600 B per buffer (on+off planes)

typedef __attribute__((ext_vector_type(2))) float f2;
typedef __attribute__((ext_vector_type(4))) unsigned int u32x4;
typedef __attribute__((ext_vector_type(8))) unsigned int u32x8;

static __device__ __forceinline__ f2 pk_fma(f2 a, f2 b, f2 c) {
  return __builtin_elementwise_fma(a, b, c);
}

// ---- gfx1250 TDM: 2D tile load, global -> LDS (CDNA5 ISA ch.7/8, TENSORcnt)
// Tile: CHUNK rows x 32 floats, row stride 256 floats in memory.
// LDS pad: after every 32 DWORDs written, skip 32 DWORDs -> on/off planes
// interleave at 128B (one 256B record per timestep).
__device__ __forceinline__ void tdm_load_tile(const void* gbase, size_t gbyte,
                                              unsigned lds_byte_addr) {
  const unsigned long long ga = (unsigned long long)gbase + gbyte;
  u32x4 g0;
  g0.s0 = 1u;                                   // count=1, user descriptor
  g0.s1 = lds_byte_addr;                        // LDS dest (bytes)
  g0.s2 = (unsigned)ga;                         // global_addr[31:0]
  g0.s3 = (unsigned)((ga >> 32) & 0x01FFFFFFu)  // global_addr[56:32]
          | (2u << 30);                         // type=2 ("image")
  u32x8 g1;
  g1.s0 = (2u << 16)                            // data_size = 4 bytes
        | (1u << 20)                            // pad_enable
        | (4u << 22)                            // pad_interval = 32 DWORDs
        | (31u << 25);                          // pad_amount  = 32 DWORDs
  g1.s1 = 0u;                                   // abar=0, tensor_dim0[15:0]=0
  g1.s2 = 0x4000u;                              // tensor_dim0 = 2^30 (no OOB)
  g1.s3 = (32u << 16) | 0x4000u;                // tile_dim0=32, tensor_dim1=2^30
  g1.s4 = (unsigned)CHUNK;                      // tile_dim1=CHUNK, tile_dim2=0
  g1.s5 = (unsigned)BATCH;                      // tensor_dim0_stride = 256 elems
  g1.s6 = 0u;
  g1.s7 = 0u;
  // 2D tensor: descriptor groups 2/3 omitted (NULL)
  asm volatile("tensor_load_to_lds %0, %1" :: "s"(g0), "s"(g1) : "memory");
}
__device__ __forceinline__ void wait_tensor0() {
  asm volatile("s_wait_tensorcnt 0" ::: "memory");
}

__global__ void __launch_bounds__(32)
lif_scan_kernel(const float* __restrict__ gic_on,
                const float* __restrict__ gic_off,
                const float* __restrict__ gsyn,
                float* __restrict__ out)
{
  // [buf][step][plane: 0=on 1=off][lane]; planes 128B apart so the two
  // per-step reads merge into one ds_load_2addr_b32.
  __shared__ __align__(16) float L[2][CHUNK][2][32];

  const int lane = threadIdx.x & 31;
  const int b0   = blockIdx.x << 5;
  const int b    = b0 + lane;

  // gsyn is a runtime input (12 uniform floats) -> scalar loads into regs
  float g[12];
#pragma unroll
  for (int ci = 0; ci < 12; ++ci) g[ci] = gsyn[ci];

  // ---------------- packed model constants (DT=0.1, CM=0.1, EK=-80) --------
  // pop pairs: q0=(On,Off) q1=(R1On,R1Off) q2=(S1,R2On) q3=(R2Off,S2)
  const f2 ELq[4] = {{-65.f,-65.f},{-65.f,-65.f},{-57.f,-65.f},{-65.f,-57.f}};
  const f2 GLq[4] = {{.005f,.005f},{.005f,.005f},{.01f,.005f},{.005f,.01f}};
  const f2 VRq[4] = {{-54.f,-54.f},{-54.f,-54.f},{-52.f,-54.f},{-54.f,-52.f}};
  const f2 TRq[4] = {{1.f,1.f},{1.f,1.f},{.5f,1.f},{1.f,.5f}};
  // connection pairs (lo,hi): P0=(c0,c4) P1=(c2,c3) P2=(c1,c6) P3=(c5,c8)
  //                           P4=(c9,c7) P5=(c10,c11)
  // P0,P1 -> pops (2,3); P2,P3 -> (4,5); P4,P5 -> (6,7)
  constexpr float A7 = 0.857142857f;    // 1 - DT/0.7
  constexpr float B15 = 0.066666667f;   // DT/1.5
  const f2 ARp[6] = {{A7,A7},{.8f,.8f},{0.f,A7},{0.f,.8f},{.8f,0.f},{A7,0.f}};
  const f2 BDp[6] = {{B15,B15},{.05f,.05f},{.1f,B15},{.1f,.05f},{.05f,.1f},{B15,.1f}};
  const f2 ESp[6] = {{0.f,0.f},{-80.f,-80.f},{0.f,0.f},{0.f,-80.f},{-80.f,0.f},{0.f,0.f}};

  // ---------------- per-lane state in VGPRs ----------------
  f2 Vq[4]; f2 tq[4]; f2 gadq[4];
#pragma unroll
  for (int q = 0; q < 4; ++q) { Vq[q] = ELq[q]; tq[q] = (f2){0.f,0.f}; gadq[q] = (f2){0.f,0.f}; }
  f2 xP[6], sP[6];                      // g-scaled synapse state (g*x, g*s)
#pragma unroll
  for (int i = 0; i < 6; ++i) { xP[i] = (f2){0.f,0.f}; sP[i] = (f2){0.f,0.f}; }

  // LDS byte addresses (low 32 bits of the shared aperture pointer)
  const unsigned lds0 = (unsigned)(uintptr_t)(&L[0][0][0][0]);
  const size_t   gcol = (size_t)b0 * 4u;   // this wave's 128B column slice

  // prefetch chunk 0 into buffer 0 (two TDM tile loads: on & off planes)
  tdm_load_tile(gic_on,  gcol, lds0);
  tdm_load_tile(gic_off, gcol, lds0 + 128u);
  wait_tensor0();

  int buf = 0;
  for (int c = 0; c < NCHUNK; ++c) {
    // prefetch chunk c+1 into the other buffer while we compute chunk c
    if (c + 1 < NCHUNK) {
      const size_t   goff = (size_t)(c + 1) * (CHUNK * BATCH * 4u) + gcol;
      const unsigned lbuf = lds0 + (unsigned)(buf ^ 1) * (unsigned)BUF_BYTES;
      tdm_load_tile(gic_on,  goff, lbuf);
      tdm_load_tile(gic_off, goff, lbuf + 128u);
    }

    // -------- process CHUNK timesteps (inputs already resident in LDS) -----
    for (int k = 0; k < CHUNK; ++k) {
      f2 gio;
      gio.x = L[buf][k][0][lane];       // gic_on[t][b]
      gio.y = L[buf][k][1][lane];       // gic_off[t][b]

      // negated synaptic currents, packed per pop pair (scatter is 2 FMAs)
      const f2 acc0 = (gio * Vq[0]) * -0.17f;                    // IC drive
      f2 acc1 = pk_fma(-sP[0], Vq[1] - ESp[0],
                pk_fma(-sP[1], Vq[1] - ESp[1], (f2){0.f,0.f}));
      f2 acc2 = pk_fma(-sP[2], Vq[2] - ESp[2],
                pk_fma(-sP[3], Vq[2] - ESp[3], (f2){0.f,0.f}));
      f2 acc3 = pk_fma(-sP[4], Vq[3] - ESp[4],
                pk_fma(-sP[5], Vq[3] - ESp[5], (f2){0.f,0.f}));

      // explicit Euler: DT/CM == 1, so Vn = V + (-GL*(V-EL) - gad*(V+80) - I)
      f2 Vc[4];
      Vc[0] = Vq[0] + pk_fma(-GLq[0], Vq[0] - ELq[0], acc0);
      Vc[1] = Vq[1] + pk_fma(-GLq[1], Vq[1] - ELq[1],
                             pk_fma(-gadq[1], Vq[1] + 80.f, acc1));
      Vc[2] = Vq[2] + pk_fma(-GLq[2], Vq[2] - ELq[2],
                             pk_fma(-gadq[2], Vq[2] + 80.f, acc2));
      Vc[3] = Vq[3] + pk_fma(-GLq[3], Vq[3] - ELq[3],
                             pk_fma(-gadq[3], Vq[3] + 80.f, acc3));

      // refractory hold, rising-edge spike, reset, refractory timer (scalar;
      // compares/selects have no packed-f32 form, VOPD dual-issues them)
      bool sp[8];
      auto neuron = [](float Vold, float Vcomp, float trold, float VRv, float TRv,
                       float& Vout, float& trn, bool& spb) {
        const bool  refr = trold > 0.f;
        const float Vn   = refr ? VRv : Vcomp;
        spb  = (Vn >= -47.f) && (Vold < -47.f);
        Vout = spb ? VRv : Vn;
        trn  = spb ? TRv : fmaxf(trold - 0.1f, 0.f);
      };
#pragma unroll
      for (int q = 0; q < 4; ++q) {
        float vx, vy, tx, ty;
        neuron(Vq[q].x, Vc[q].x, tq[q].x, VRq[q].x, TRq[q].x, vx, tx, sp[2*q]);
        neuron(Vq[q].y, Vc[q].y, tq[q].y, VRq[q].y, TRq[q].y, vy, ty, sp[2*q+1]);
        Vq[q] = (f2){vx, vy};
        tq[q] = (f2){tx, ty};
      }

      // adaptation (pops 2,3,5,6 only; dead halves stay exactly 0)
      gadq[1] = pk_fma(gadq[1], (f2){0.999f,0.999f},
                       (f2){sp[2] ? 3e-4f : 0.f, sp[3] ? 3e-4f : 0.f});
      gadq[2] = pk_fma(gadq[2], (f2){0.f,0.999f},
                       (f2){0.f, sp[5] ? 3e-4f : 0.f});
      gadq[3] = pk_fma(gadq[3], (f2){0.999f,0.f},
                       (f2){sp[6] ? 3e-4f : 0.f, 0.f});

      // double-exponential PSCs, g folded into state: kick = sp ? gsyn : 0
      const f2 K[6] = {
        {sp[0] ? g[0]  : 0.f, sp[1] ? g[4]  : 0.f},   // P0=(c0,c4) src(0,1)
        {sp[4] ? g[2]  : 0.f, sp[4] ? g[3]  : 0.f},   // P1=(c2,c3) src(4,4)
        {sp[0] ? g[1]  : 0.f, sp[2] ? g[6]  : 0.f},   // P2=(c1,c6) src(0,2)
        {sp[1] ? g[5]  : 0.f, sp[7] ? g[8]  : 0.f},   // P3=(c5,c8) src(1,7)
        {sp[7] ? g[9]  : 0.f, sp[2] ? g[7]  : 0.f},   // P4=(c9,c7) src(7,2)
        {sp[3] ? g[10] : 0.f, sp[3] ? g[11] : 0.f},   // P5=(c10,c11) src(3,3)
      };
#pragma unroll
      for (int i = 0; i < 6; ++i) {
        const f2 xn = pk_fma(xP[i], ARp[i], K[i]);
        xP[i] = xn;
        sP[i] = pk_fma(BDp[i], xn - sP[i], sP[i]);
      }

      // emit V of R2On (pop5 = q2.y), R2Off (pop6 = q3.x): [T,B,2]
      const size_t t = (size_t)c * CHUNK + (size_t)k;
      reinterpret_cast<float2*>(out)[t * BATCH + b] =
          make_float2(Vq[2].y, Vq[3].x);
    }

    wait_tensor0();   // chunk c+1 (if any) has landed in the other buffer
    buf ^= 1;
  }
}

extern "C" void kernel_launch(void* const* d_in, const int* in_sizes, int n_in,
                              void* d_out, int out_size, void* d_ws, size_t ws_size,
                              hipStream_t stream) {
  (void)in_sizes; (void)n_in; (void)out_size; (void)d_ws; (void)ws_size;
  const float* gic_on  = (const float*)d_in[0];
  const float* gic_off = (const float*)d_in[1];
  const float* gsyn    = (const float*)d_in[2];
  float* out = (float*)d_out;

  // 256 batch elements -> 8 single-wave (32-thread) workgroups, one per WGP,
  // so each wave owns a SIMD32 for the latency-bound sequential scan.
  lif_scan_kernel<<<dim3(BATCH / 32), dim3(32), 0, stream>>>(gic_on, gic_off,
                                                             gsyn, out);
}